// Transmission_Layer_88244398063844
// MI455X (gfx1250) — compile-verified
//
#include <hip/hip_runtime.h>

typedef __attribute__((ext_vector_type(2))) float v2f;
typedef __attribute__((ext_vector_type(8))) float v8f;

constexpr int kB   = 16384;
constexpr int kNr  = 32;
constexpr int kS   = 32;
constexpr int kQ   = 5;
constexpr int kNRF = 4;
constexpr int kMr  = kQ * kNRF;   // 20
constexpr int kD   = kMr * kS;    // 640
constexpr int kN2  = 2 * kD;      // 1280 (real/imag expanded GEMM dim)

// ---------------------------------------------------------------------------
// Kernel A: expand conj(U)^T (complex, D x D) into a real N2 x N2 matrix so
// that out[b, m2] = sum_k2 A2[b, k2] * Ubig[k2, m2] realizes U^H y with the
// output already in the reference's [D,2] (real,imag) interleaved layout.
//   out_r[m] = sum_k Ur[k,m]*yr[k] + Ui[k,m]*yi[k]
//   out_i[m] = sum_k Ur[k,m]*yi[k] - Ui[k,m]*yr[k]
// ---------------------------------------------------------------------------
__global__ void build_Ubig(const float* __restrict__ U_ri,
                           float* __restrict__ Ubig) {
    int gid = blockIdx.x * blockDim.x + threadIdx.x;   // 640*640 threads
    if (gid >= kD * kD) return;
    int k = gid / kD;
    int m = gid - k * kD;
    float ur = U_ri[(k * kD + m) * 2 + 0];
    float ui = U_ri[(k * kD + m) * 2 + 1];
    Ubig[(size_t)(2 * k    ) * kN2 + 2 * m    ] =  ur;
    Ubig[(size_t)(2 * k    ) * kN2 + 2 * m + 1] = -ui;
    Ubig[(size_t)(2 * k + 1) * kN2 + 2 * m    ] =  ui;
    Ubig[(size_t)(2 * k + 1) * kN2 + 2 * m + 1] =  ur;
}

// ---------------------------------------------------------------------------
// Kernel B: fused noise-add + W combine + pre-whiten + transpose.
// One thread per (b, s) column. Produces A2[b][2*(s*Mr+m)+{re,im}].
// Memory bound: streams 671MB noise + 134MB H exactly once.
// ---------------------------------------------------------------------------
__global__ __launch_bounds__(256)
void build_A2(const float* __restrict__ H,
              const float* __restrict__ noise,
              const float* __restrict__ PW,
              const float* __restrict__ W,
              float* __restrict__ A2) {
    __shared__ float Wsh[kNr * kMr];        // 640 floats
    __shared__ float PWsh[kMr * kMr * 2];   // 800 floats
    const int tid = threadIdx.x;
    for (int i = tid; i < kNr * kMr; i += blockDim.x)      Wsh[i]  = W[i];
    for (int i = tid; i < kMr * kMr * 2; i += blockDim.x)  PWsh[i] = PW[i];
    __syncthreads();

    const int gid = blockIdx.x * blockDim.x + tid;  // kB * kS threads
    const int b = gid >> 5;
    const int s = gid & 31;

    // H column s for this batch element (kept in registers, reused for all q)
    float hr[kNr], hi[kNr];
#pragma unroll
    for (int n = 0; n < kNr; ++n) {
        size_t idx = ((size_t)(b * kNr + n) * kS + s) * 2;
        hr[n] = H[idx + 0];
        hi[n] = H[idx + 1];
    }

    float y1r[kMr], y1i[kMr];
    for (int q = 0; q < kQ; ++q) {
        float ar[kNRF] = {0.f, 0.f, 0.f, 0.f};
        float ai[kNRF] = {0.f, 0.f, 0.f, 0.f};
        for (int n = 0; n < kNr; ++n) {
            size_t idx = (((size_t)(q * kB + b) * kNr + n) * kS + s) * 2;
            float zr = hr[n] + noise[idx + 0];
            float zi = hi[n] + noise[idx + 1];
#pragma unroll
            for (int r = 0; r < kNRF; ++r) {
                float w = Wsh[n * kMr + q * kNRF + r];   // W is real
                ar[r] = fmaf(w, zr, ar[r]);
                ai[r] = fmaf(w, zi, ai[r]);
            }
        }
#pragma unroll
        for (int r = 0; r < kNRF; ++r) {
            y1r[q * kNRF + r] = ar[r];
            y1i[q * kNRF + r] = ai[r];
        }
    }

    // pre-whitening: y2[m] = sum_k PW[m][k] * y1[k]  (complex x complex)
    float* outp = &A2[(size_t)b * kN2];
#pragma unroll
    for (int m = 0; m < kMr; ++m) {
        float sr = 0.f, si = 0.f;
#pragma unroll
        for (int k = 0; k < kMr; ++k) {
            float pr = PWsh[(m * kMr + k) * 2 + 0];
            float pi = PWsh[(m * kMr + k) * 2 + 1];
            sr = fmaf(pr, y1r[k], fmaf(-pi, y1i[k], sr));
            si = fmaf(pr, y1i[k], fmaf( pi, y1r[k], si));
        }
        outp[2 * (s * kMr + m) + 0] = sr;   // y[b, s*Mr + m] layout (transpose)
        outp[2 * (s * kMr + m) + 1] = si;
    }
}

// ---------------------------------------------------------------------------
// Kernel C: fp32 WMMA GEMM  C[16384 x 1280] = A2 * Ubig, written to d_out.
// Grid: x = N blocks (10)  -> concurrent blocks share A row-panels via L2,
//       y = M blocks (128) -> Ubig (6.5MB) stays L2-resident.
// 256 threads = 8 waves; 128x128 block tile; each wave: 64x32 (4x2 frags).
// Software-pipelined: next k-tile staged to registers while WMMAs consume
// the current LDS tile; prefetch 2 tiles ahead (global_prefetch_b8).
// ---------------------------------------------------------------------------
constexpr int kAPad = 36;   // 16B-aligned rows; m*36 % 64 -> 16 distinct banks

__global__ __launch_bounds__(256)
void gemm_wmma(const float* __restrict__ A2,
               const float* __restrict__ Ubig,
               float* __restrict__ Cout) {
    __shared__ float As[128][kAPad];  // 128 rows x 32 k (padded)
    __shared__ float Bs[32][128];     // 32 k x 128 n

    const int tid  = threadIdx.x;
    const int bn   = blockIdx.x * 128;   // N dimension (10 blocks)
    const int bm   = blockIdx.y * 128;   // M dimension (128 blocks)

    const int wid    = tid >> 5;
    const int lane   = tid & 31;
    const int lane16 = lane & 15;
    const int khalf  = (lane >> 4) << 1;        // lanes 0-15: K=0,1 ; 16-31: K=2,3
    const int wm     = (wid & 1) * 64;
    const int wn     = (wid >> 1) * 32;

    // per-thread staging coordinates (4 float4 each for A and B tiles)
    int arow[4], akq[4], bkk[4], bnq[4];
#pragma unroll
    for (int i = 0; i < 4; ++i) {
        int linear = tid + i * 256;          // 0..1023 float4 slots
        arow[i] = linear >> 3;               // tile row   (A: 128 x 8 slots)
        akq[i]  = (linear & 7) << 2;         // k within tile
        bkk[i]  = linear >> 5;               // tile k row (B: 32 x 32 slots)
        bnq[i]  = (linear & 31) << 2;        // n within tile
    }

    const v8f vzero = {0.f, 0.f, 0.f, 0.f, 0.f, 0.f, 0.f, 0.f};
    v8f acc[4][2];
#pragma unroll
    for (int i = 0; i < 4; ++i)
#pragma unroll
        for (int j = 0; j < 2; ++j) acc[i][j] = vzero;

    float4 ra[4], rb[4];
    // prologue: stage k-tile 0 into registers
#pragma unroll
    for (int i = 0; i < 4; ++i) {
        ra[i] = *reinterpret_cast<const float4*>(
            &A2[(size_t)(bm + arow[i]) * kN2 + akq[i]]);
        rb[i] = *reinterpret_cast<const float4*>(
            &Ubig[(size_t)bkk[i] * kN2 + bn + bnq[i]]);
    }

    for (int k0 = 0; k0 < kN2; k0 += 32) {
        // commit staged registers to LDS (hardware waits loadcnt here)
#pragma unroll
        for (int i = 0; i < 4; ++i) {
            *reinterpret_cast<float4*>(&As[arow[i]][akq[i]]) = ra[i];
            *reinterpret_cast<float4*>(&Bs[bkk[i]][bnq[i]])  = rb[i];
        }
        __syncthreads();

        const int kn = k0 + 32;
        if (kn < kN2) {
            // stage next k-tile; loads fly while WMMAs below execute
#pragma unroll
            for (int i = 0; i < 4; ++i) {
                ra[i] = *reinterpret_cast<const float4*>(
                    &A2[(size_t)(bm + arow[i]) * kN2 + kn + akq[i]]);
                rb[i] = *reinterpret_cast<const float4*>(
                    &Ubig[(size_t)(kn + bkk[i]) * kN2 + bn + bnq[i]]);
            }
            // hint L2 two tiles ahead (global_prefetch_b8)
            if (kn + 32 < kN2) {
                __builtin_prefetch(&A2[(size_t)(bm + arow[0]) * kN2 + kn + 32 + akq[0]], 0, 1);
                __builtin_prefetch(&Ubig[(size_t)(kn + 32 + bkk[0]) * kN2 + bn + bnq[0]], 0, 1);
            }
        }

#pragma unroll
        for (int kt = 0; kt < 32; kt += 4) {
            v2f afrag[4], bfrag[2];
#pragma unroll
            for (int i = 0; i < 4; ++i) {
                int m = wm + i * 16 + lane16;
                afrag[i].x = As[m][kt + khalf];
                afrag[i].y = As[m][kt + khalf + 1];
            }
#pragma unroll
            for (int j = 0; j < 2; ++j) {
                int n = wn + j * 16 + lane16;
                bfrag[j].x = Bs[kt + khalf][n];
                bfrag[j].y = Bs[kt + khalf + 1][n];
            }
#pragma unroll
            for (int i = 0; i < 4; ++i)
#pragma unroll
                for (int j = 0; j < 2; ++j)
                    acc[i][j] = __builtin_amdgcn_wmma_f32_16x16x4_f32(
                        false, afrag[i], false, bfrag[j],
                        (short)0, acc[i][j], false, false);
        }
        __syncthreads();
    }

    // epilogue: C layout — VGPR v: lanes 0-15 -> M=v, lanes 16-31 -> M=v+8
    const int rbase = (lane >> 4) << 3;
#pragma unroll
    for (int i = 0; i < 4; ++i) {
#pragma unroll
        for (int j = 0; j < 2; ++j) {
            int col = bn + wn + j * 16 + lane16;
#pragma unroll
            for (int v = 0; v < 8; ++v) {
                int row = bm + wm + i * 16 + rbase + v;
                Cout[(size_t)row * kN2 + col] = acc[i][j][v];
            }
        }
    }
}

// ---------------------------------------------------------------------------
extern "C" void kernel_launch(void* const* d_in, const int* in_sizes, int n_in,
                              void* d_out, int out_size, void* d_ws, size_t ws_size,
                              hipStream_t stream) {
    const float* H     = (const float*)d_in[0];  // [B, Nr, S, 2]
    const float* noise = (const float*)d_in[1];  // [Q, B, Nr, S, 2]
    const float* PW    = (const float*)d_in[2];  // [Mr, Mr, 2]
    const float* U     = (const float*)d_in[3];  // [D, D, 2]
    const float* W     = (const float*)d_in[4];  // [Nr, Mr]
    float* out = (float*)d_out;                  // [B, D, 2] == [B, 1280]

    float* A2   = (float*)d_ws;                                             // 83.9 MB
    float* Ubig = (float*)((char*)d_ws + (size_t)kB * kN2 * sizeof(float)); // 6.6 MB

    build_Ubig<<<(kD * kD + 255) / 256, 256, 0, stream>>>(U, Ubig);
    build_A2<<<(kB * kS) / 256, 256, 0, stream>>>(H, noise, PW, W, A2);

    dim3 grid(kN2 / 128, kB / 128);   // x = N (10), y = M (128): L2-friendly
    gemm_wmma<<<grid, 256, 0, stream>>>(A2, Ubig, out);
}